// EfficientMultiTensorProduct_28217935135338
// MI455X (gfx1250) — compile-verified
//
#include <hip/hip_runtime.h>
#include <stdint.h>

typedef __attribute__((ext_vector_type(2))) float v2f;
typedef __attribute__((ext_vector_type(4))) float v4f;
typedef __attribute__((ext_vector_type(8))) float v8f;

constexpr int kNF      = 952;          // 136 pair + 816 triple features
constexpr int kNPair   = 136;
constexpr int kK2Steps = kNPair / 4;   // 34
constexpr int kKSteps  = kNF / 4;      // 238
constexpr int kC       = 128;
constexpr int kItemsPerBlock = 128;    // 4 waves x 32 items
constexpr int kZStride = 156;          // 136 z2 + 16 x + 1.0 + pad (16B aligned, banks coprime)
constexpr int kXBase   = 136;          // x[] lives at row offset 136..151
constexpr int kOneSlot = 152;          // constant 1.0

// input coefficient index a (0..15) -> (l, grid column m in 0..6)
__device__ __forceinline__ void amap(int a, int& l, int& m) {
  if (a == 0)      { l = 0; m = 3; }
  else if (a < 4)  { l = 1; m = a + 1; }
  else if (a < 9)  { l = 2; m = a - 3; }
  else             { l = 3; m = a - 9; }
}

// ---------------- setup kernel 1: monomial tables ----------------
// ftab : (i | j<<8 | k<<16) monomial indices, k==16 means "times 1.0" (pair)
// ftab2: packed LDS word-offsets for the main loop: pairIdx | ((kXBase+c)<<16)
__global__ void build_ftab_kernel(unsigned* __restrict__ ftab, unsigned* __restrict__ ftab2) {
  if (blockIdx.x == 0 && threadIdx.x == 0) {
    int f = 0;
    for (int a = 0; a < 16; ++a)
      for (int b = a; b < 16; ++b) {
        ftab[f]  = (unsigned)a | ((unsigned)b << 8) | (16u << 16);
        ftab2[f] = (unsigned)f | ((unsigned)kOneSlot << 16);
        ++f;
      }
    int pidx = 0;
    for (int a = 0; a < 16; ++a)
      for (int b = a; b < 16; ++b) {
        for (int c = b; c < 16; ++c) {
          ftab[f]  = (unsigned)a | ((unsigned)b << 8) | ((unsigned)c << 16);
          ftab2[f] = (unsigned)pidx | ((unsigned)(kXBase + c) << 16);
          ++f;
        }
        ++pidx;
      }
  }
}

// ---------------- setup kernel 2: W[952][16] coefficient matrix ----------------
__global__ void build_w_kernel(const unsigned* __restrict__ ftab,
                               const float* __restrict__ b1re, const float* __restrict__ b1im,
                               const float* __restrict__ b2re, const float* __restrict__ b2im,
                               const float* __restrict__ b3re, const float* __restrict__ b3im,
                               float* __restrict__ W) {
  int gid = blockIdx.x * blockDim.x + threadIdx.x;
  if (gid >= kNF * 16) return;
  int f = gid >> 4, k = gid & 15;
  unsigned t = ftab[f];
  int i = t & 255, j = (t >> 8) & 255, kk = (t >> 16) & 255;
  int li, mi, lj, mj, lo, mo;
  amap(i, li, mi); amap(j, lj, mj); amap(k, lo, mo);

  // g = B1[i] (*) B1[j]   (complex 1-D conv over u, length 13)
  float gre[13], gim[13];
  for (int p = 0; p < 13; ++p) { gre[p] = 0.f; gim[p] = 0.f; }
  const float* aire = b1re + li * 49 + mi * 7;
  const float* aiim = b1im + li * 49 + mi * 7;
  const float* ajre = b1re + lj * 49 + mj * 7;
  const float* ajim = b1im + lj * 49 + mj * 7;
  for (int u = 0; u < 7; ++u)
    for (int v = 0; v < 7; ++v) {
      float ar = aire[u], ai = aiim[u], br = ajre[v], bi = ajim[v];
      gre[u + v] += ar * br - ai * bi;
      gim[u + v] += ar * bi + ai * br;
    }

  float res = 0.0f;
  if (kk == 16) {                       // quadratic feature
    if (mi + mj == mo + 3) {
      const float* cre = b2re + lo * 169 + (mo + 3) * 13;
      const float* cim = b2im + lo * 169 + (mo + 3) * 13;
      float acc = 0.0f;
      for (int p = 0; p < 13; ++p) acc += gre[p] * cre[p] - gim[p] * cim[p];
      res = (i == j ? 1.0f : 2.0f) * acc;
    }
  } else {                              // cubic feature
    int lk3, mk3; amap(kk, lk3, mk3);
    if (mi + mj + mk3 == mo + 6) {
      const float* akre = b1re + lk3 * 49 + mk3 * 7;
      const float* akim = b1im + lk3 * 49 + mk3 * 7;
      float hre[19], him[19];
      for (int q = 0; q < 19; ++q) { hre[q] = 0.f; him[q] = 0.f; }
      for (int p = 0; p < 13; ++p)
        for (int w = 0; w < 7; ++w) {
          float ar = gre[p], ai = gim[p], br = akre[w], bi = akim[w];
          hre[p + w] += ar * br - ai * bi;
          him[p + w] += ar * bi + ai * br;
        }
      const float* cre = b3re + lo * 361 + (mo + 6) * 19;
      const float* cim = b3im + lo * 361 + (mo + 6) * 19;
      float acc = 0.0f;
      for (int q = 0; q < 19; ++q) acc += hre[q] * cre[q] - him[q] * cim[q];
      float mult = (i == j && j == kk) ? 1.0f : ((i == j || j == kk) ? 3.0f : 6.0f);
      res = mult * acc;
    }
  }
  W[f * 16 + k] = res;
}

// ---------------- main fused kernel ----------------
__launch_bounds__(128)
__global__ void emtp_main_kernel(const float* __restrict__ atom_feat,
                                 const float* __restrict__ atom_type,
                                 const float* __restrict__ w1,
                                 const float* __restrict__ w2,
                                 const float* __restrict__ w3,
                                 const unsigned* __restrict__ ftab2_g,
                                 const float* __restrict__ W_g,
                                 float* __restrict__ out) {
  extern __shared__ char smem_raw[];
  float* zrowbase = (float*)smem_raw;                        // 128 * kZStride
  float* mixl = zrowbase + kItemsPerBlock * kZStride;        // 128 * 12

  const int tid = threadIdx.x;
  const int blockBase = blockIdx.x * kItemsPerBlock;

  { // phase 1: one item per thread -> z2 products, x copy, mixing weights
    int gItem = blockBase + tid;
    const v4f* xp = (const v4f*)(atom_feat + (size_t)gItem * 16);
    v4f q0 = xp[0], q1 = xp[1], q2 = xp[2], q3 = xp[3];
    float x[16];
#pragma unroll
    for (int a = 0; a < 4; ++a) { x[a] = q0[a]; x[4 + a] = q1[a]; x[8 + a] = q2[a]; x[12 + a] = q3[a]; }
    float* zrow = zrowbase + tid * kZStride;
    int f = 0;
#pragma unroll
    for (int a = 0; a < 16; ++a)
#pragma unroll
      for (int b = a; b < 16; ++b) { zrow[f] = x[a] * x[b]; ++f; }
#pragma unroll
    for (int a = 0; a < 16; ++a) zrow[kXBase + a] = x[a];
    zrow[kOneSlot] = 1.0f;

    int n = gItem >> 7, c = gItem & 127;
    float at0 = atom_type[n * 4 + 0], at1 = atom_type[n * 4 + 1];
    float at2 = atom_type[n * 4 + 2], at3 = atom_type[n * 4 + 3];
#pragma unroll
    for (int l = 0; l < 4; ++l) {
      int wi = c * 4 + l;
      mixl[tid * 12 + l]     = at0 * w1[wi] + at1 * w1[512 + wi] + at2 * w1[1024 + wi] + at3 * w1[1536 + wi];
      mixl[tid * 12 + 4 + l] = at0 * w2[wi] + at1 * w2[512 + wi] + at2 * w2[1024 + wi] + at3 * w2[1536 + wi];
      mixl[tid * 12 + 8 + l] = at0 * w3[wi] + at1 * w3[512 + wi] + at2 * w3[1024 + wi] + at3 * w3[1536 + wi];
    }
  }
  __syncthreads();

  const int lane = tid & 31;
  const int wid  = tid >> 5;
  const int half = lane >> 4;       // A/B fragment half: K+0/1 vs K+2/3
  const int row  = lane & 15;       // A: M row; B/C/D: N column
  const float* z0 = zrowbase + (wid * 32 + row) * kZStride;        // M-tile 0 item row
  const float* z1 = zrowbase + (wid * 32 + 16 + row) * kZStride;   // M-tile 1 item row
  const float* Wlane = W_g + 2 * half * 16 + row;                  // + 64*s per step, +16 for K+1
  const uint2* tpack = (const uint2*)ftab2_g;                      // [s*2+half] = offsets for f0,f0+1

  v8f acc2a = {}, acc2b = {}, acc3a = {}, acc3b = {};

  // ---- pair region: A fragments are the staged z2 values themselves ----
  for (int s = 0; s < kK2Steps; ++s) {
    int f0 = s * 4 + half * 2;
    v2f A0 = *(const v2f*)(z0 + f0);
    v2f A1 = *(const v2f*)(z1 + f0);
    v2f B;
    B.x = Wlane[s * 64];
    B.y = Wlane[s * 64 + 16];
    acc2a = __builtin_amdgcn_wmma_f32_16x16x4_f32(false, A0, false, B, (short)0, acc2a, false, false);
    acc2b = __builtin_amdgcn_wmma_f32_16x16x4_f32(false, A1, false, B, (short)0, acc2b, false, false);
  }

  // ---- triple region: z3 = z2[pair] * x[c], offsets pre-packed ----
#pragma unroll 2
  for (int s = kK2Steps; s < kKSteps; ++s) {
    uint2 t = tpack[s * 2 + half];
    int p0 = t.x & 0xffff, c0 = t.x >> 16;
    int p1 = t.y & 0xffff, c1 = t.y >> 16;
    v2f A0, A1, B;
    A0.x = z0[p0] * z0[c0];
    A0.y = z0[p1] * z0[c1];
    A1.x = z1[p0] * z1[c0];
    A1.y = z1[p1] * z1[c1];
    B.x = Wlane[s * 64];
    B.y = Wlane[s * 64 + 16];
    acc3a = __builtin_amdgcn_wmma_f32_16x16x4_f32(false, A0, false, B, (short)0, acc3a, false, false);
    acc3b = __builtin_amdgcn_wmma_f32_16x16x4_f32(false, A1, false, B, (short)0, acc3b, false, false);
  }

  // ---- epilogue: out = mix1*x + mix2*y2 + mix3*y3, per-degree scattered layout ----
  int kcol = row;
  int lk, mk; amap(kcol, lk, mk);
  const int width = 2 * lk + 1;
  const int base_off = kC * lk * lk;          // {0,128,512,1152}
  const int pos = mk - 3 + lk;
#pragma unroll
  for (int r = 0; r < 8; ++r) {
    {
      int it = wid * 32 + r + 8 * half;
      float val = mixl[it * 12 + lk] * zrowbase[it * kZStride + kXBase + kcol]
                + mixl[it * 12 + 4 + lk] * acc2a[r]
                + mixl[it * 12 + 8 + lk] * acc3a[r];
      int gItem = blockBase + it;
      int n = gItem >> 7, c = gItem & 127;
      out[n * 2048 + base_off + c * width + pos] = val;
    }
    {
      int it = wid * 32 + 16 + r + 8 * half;
      float val = mixl[it * 12 + lk] * zrowbase[it * kZStride + kXBase + kcol]
                + mixl[it * 12 + 4 + lk] * acc2b[r]
                + mixl[it * 12 + 8 + lk] * acc3b[r];
      int gItem = blockBase + it;
      int n = gItem >> 7, c = gItem & 127;
      out[n * 2048 + base_off + c * width + pos] = val;
    }
  }
}

extern "C" void kernel_launch(void* const* d_in, const int* in_sizes, int n_in,
                              void* d_out, int out_size, void* d_ws, size_t ws_size,
                              hipStream_t stream) {
  const float* atom_feat = (const float*)d_in[0];
  const float* atom_type = (const float*)d_in[1];
  const float* w1 = (const float*)d_in[2];
  const float* w2 = (const float*)d_in[3];
  const float* w3 = (const float*)d_in[4];
  const float* b1re = (const float*)d_in[5];
  const float* b1im = (const float*)d_in[6];
  const float* b2re = (const float*)d_in[7];
  const float* b2im = (const float*)d_in[8];
  const float* b3re = (const float*)d_in[9];
  const float* b3im = (const float*)d_in[10];
  float* out = (float*)d_out;

  unsigned* ftab  = (unsigned*)d_ws;                     // 952 u32
  unsigned* ftab2 = (unsigned*)((char*)d_ws + 4096);     // 952 u32 packed offsets
  float*    W     = (float*)((char*)d_ws + 8192);        // 952*16 f32

  build_ftab_kernel<<<1, 32, 0, stream>>>(ftab, ftab2);
  build_w_kernel<<<(kNF * 16 + 255) / 256, 256, 0, stream>>>(ftab, b1re, b1im, b2re, b2im, b3re, b3im, W);

  size_t lds_bytes = (size_t)(kItemsPerBlock * kZStride + kItemsPerBlock * 12) * sizeof(float); // ~86 KB
  int nBlocks = (1024 * kC) / kItemsPerBlock;   // 1024 blocks
  emtp_main_kernel<<<nBlocks, 128, lds_bytes, stream>>>(atom_feat, atom_type, w1, w2, w3, ftab2, W, out);
}